// GAT_Net_85985245266267
// MI455X (gfx1250) — compile-verified
//
#include <hip/hip_runtime.h>
#include <math.h>

#define N_NODES 100000
#define N_EDGES 1600000
#define F_IN    128
#define HID     8
#define HEADS   8
#define F1      (HEADS * HID)   // 64
#define N_CLS   16
#define NEG_SLOPE 0.2f

typedef float v2f __attribute__((ext_vector_type(2)));
typedef float v8f __attribute__((ext_vector_type(8)));

__device__ __forceinline__ float leaky(float x) { return x >= 0.f ? x : NEG_SLOPE * x; }

// float atomic max via signed/unsigned int monotonicity trick (init must be -inf)
__device__ __forceinline__ void atomicMaxF(float* addr, float v) {
    if (v >= 0.f) atomicMax((int*)addr, __float_as_int(v));
    else          atomicMin((unsigned int*)addr, __float_as_uint(v));
}

// ---------------------------------------------------------------------------
// Init accumulators (must run every call; harness does not re-poison)
// ---------------------------------------------------------------------------
__global__ void init_kernel(float* m1, float* den1, float* out1,
                            float* m2, float* den2, float* out2) {
    int i = blockIdx.x * blockDim.x + threadIdx.x;
    int stride = gridDim.x * blockDim.x;
    for (int j = i; j < N_NODES * HEADS; j += stride) { m1[j] = -INFINITY; den1[j] = 0.f; }
    for (int j = i; j < N_NODES * F1;    j += stride) { out1[j] = 0.f; }
    for (int j = i; j < N_NODES;         j += stride) { m2[j] = -INFINITY; den2[j] = 0.f; }
    for (int j = i; j < N_NODES * N_CLS; j += stride) { out2[j] = 0.f; }
}

// ---------------------------------------------------------------------------
// GEMM1: h1[N,64] = x[N,128] @ W1[128,64]   (exact f32 via V_WMMA_F32_16X16X4_F32)
// block = 128 threads = 4 waves; wave w -> column tile n0 = 16*w; block -> 16 rows
// ---------------------------------------------------------------------------
__global__ __launch_bounds__(128) void gemm1_kernel(const float* __restrict__ x,
                                                    const float* __restrict__ W1,
                                                    float* __restrict__ h1) {
    const int wave = threadIdx.x >> 5;
    const int lane = threadIdx.x & 31;
    const int half = lane >> 4;      // 0: lanes 0-15, 1: lanes 16-31
    const int l    = lane & 15;
    const int m0   = blockIdx.x * 16;
    const int n0   = wave * 16;

    v8f c = {0.f, 0.f, 0.f, 0.f, 0.f, 0.f, 0.f, 0.f};
    const float* xrow = x + (size_t)(m0 + l) * F_IN;

    for (int k0 = 0; k0 < F_IN; k0 += 4) {
        v2f a, b;
        const int ka = k0 + 2 * half;              // this lane-half's K pair
        a.x = xrow[ka];
        a.y = xrow[ka + 1];
        b.x = W1[(size_t)ka * F1 + n0 + l];
        b.y = W1[(size_t)(ka + 1) * F1 + n0 + l];
        c = __builtin_amdgcn_wmma_f32_16x16x4_f32(false, a, false, b, (short)0, c,
                                                  false, false);
    }
#pragma unroll
    for (int i = 0; i < 8; ++i) {
        const int row = m0 + i + 8 * half;
        h1[(size_t)row * F1 + n0 + l] = c[i];
    }
}

// ---------------------------------------------------------------------------
// Per-node attention coefficients, layer 1: alpha_s/d[n,h] = <h1[n,h,:], a1[h,:]>
// ---------------------------------------------------------------------------
__global__ void alphas1_kernel(const float* __restrict__ h1,
                               const float* __restrict__ a1s,
                               const float* __restrict__ a1d,
                               float* __restrict__ as, float* __restrict__ ad) {
    int n = blockIdx.x * blockDim.x + threadIdx.x;
    if (n >= N_NODES) return;
    const float* hr = h1 + (size_t)n * F1;
#pragma unroll
    for (int h = 0; h < HEADS; ++h) {
        float ss = 0.f, sd = 0.f;
#pragma unroll
        for (int d = 0; d < HID; ++d) {
            float v = hr[h * HID + d];
            ss += v * a1s[h * HID + d];
            sd += v * a1d[h * HID + d];
        }
        as[(size_t)n * HEADS + h] = ss;
        ad[(size_t)n * HEADS + h] = sd;
    }
}

// ---------------------------------------------------------------------------
// Edge passes, layer 1 (8 heads)
// ---------------------------------------------------------------------------
__global__ void edge1_max(const int* __restrict__ src, const int* __restrict__ dst,
                          const float* __restrict__ as, const float* __restrict__ ad,
                          float* __restrict__ m1) {
    int stride = gridDim.x * blockDim.x;
    for (int e = blockIdx.x * blockDim.x + threadIdx.x; e < N_EDGES; e += stride) {
        int s = src[e], d = dst[e];
        const float4* a0 = (const float4*)(as + (size_t)s * HEADS);
        const float4* b0 = (const float4*)(ad + (size_t)d * HEADS);
        float4 av[2] = {a0[0], a0[1]};
        float4 bv[2] = {b0[0], b0[1]};
        const float* af = (const float*)av;
        const float* bf = (const float*)bv;
#pragma unroll
        for (int h = 0; h < HEADS; ++h)
            atomicMaxF(&m1[(size_t)d * HEADS + h], leaky(af[h] + bf[h]));
    }
}

__global__ void edge1_den(const int* __restrict__ src, const int* __restrict__ dst,
                          const float* __restrict__ as, const float* __restrict__ ad,
                          const float* __restrict__ m1, float* __restrict__ den1) {
    int stride = gridDim.x * blockDim.x;
    for (int e = blockIdx.x * blockDim.x + threadIdx.x; e < N_EDGES; e += stride) {
        int s = src[e], d = dst[e];
        const float4* a0 = (const float4*)(as + (size_t)s * HEADS);
        const float4* b0 = (const float4*)(ad + (size_t)d * HEADS);
        const float4* m0 = (const float4*)(m1 + (size_t)d * HEADS);
        float4 av[2] = {a0[0], a0[1]};
        float4 bv[2] = {b0[0], b0[1]};
        float4 mv[2] = {m0[0], m0[1]};
        const float* af = (const float*)av;
        const float* bf = (const float*)bv;
        const float* mf = (const float*)mv;
#pragma unroll
        for (int h = 0; h < HEADS; ++h) {
            float ex = expf(leaky(af[h] + bf[h]) - mf[h]);
            atomicAdd(&den1[(size_t)d * HEADS + h], ex);
        }
    }
}

__global__ void edge1_agg(const int* __restrict__ src, const int* __restrict__ dst,
                          const float* __restrict__ as, const float* __restrict__ ad,
                          const float* __restrict__ m1, const float* __restrict__ den1,
                          const float* __restrict__ h1, float* __restrict__ out1) {
    int stride = gridDim.x * blockDim.x;
    for (int e = blockIdx.x * blockDim.x + threadIdx.x; e < N_EDGES; e += stride) {
        int s = src[e], d = dst[e];
        int en = e + stride;
        if (en < N_EDGES)   // get next gather row in flight (global_prefetch_b8)
            __builtin_prefetch(&h1[(size_t)src[en] * F1], 0, 0);
        const float4* a0 = (const float4*)(as + (size_t)s * HEADS);
        const float4* b0 = (const float4*)(ad + (size_t)d * HEADS);
        const float4* m0 = (const float4*)(m1 + (size_t)d * HEADS);
        const float4* d0 = (const float4*)(den1 + (size_t)d * HEADS);
        float4 av[2] = {a0[0], a0[1]};
        float4 bv[2] = {b0[0], b0[1]};
        float4 mv[2] = {m0[0], m0[1]};
        float4 dv[2] = {d0[0], d0[1]};
        const float* af = (const float*)av;
        const float* bf = (const float*)bv;
        const float* mf = (const float*)mv;
        const float* df = (const float*)dv;
#pragma unroll
        for (int h = 0; h < HEADS; ++h) {
            float ex = expf(leaky(af[h] + bf[h]) - mf[h]);
            float alpha = ex / df[h];
            const float4* hv = (const float4*)(h1 + (size_t)s * F1 + h * HID);
            float4 v0 = hv[0], v1 = hv[1];
            float* o = out1 + (size_t)d * F1 + h * HID;
            atomicAdd(o + 0, alpha * v0.x);
            atomicAdd(o + 1, alpha * v0.y);
            atomicAdd(o + 2, alpha * v0.z);
            atomicAdd(o + 3, alpha * v0.w);
            atomicAdd(o + 4, alpha * v1.x);
            atomicAdd(o + 5, alpha * v1.y);
            atomicAdd(o + 6, alpha * v1.z);
            atomicAdd(o + 7, alpha * v1.w);
        }
    }
}

// out1 + b1 then ELU -> act1
__global__ void elu_kernel(const float* __restrict__ out1, const float* __restrict__ b1,
                           float* __restrict__ act1) {
    int stride = gridDim.x * blockDim.x;
    for (int i = blockIdx.x * blockDim.x + threadIdx.x; i < N_NODES * F1; i += stride) {
        float v = out1[i] + b1[i & (F1 - 1)];
        act1[i] = v > 0.f ? v : expm1f(v);
    }
}

// ---------------------------------------------------------------------------
// GEMM2: h2[N,16] = act1[N,64] @ W2[64,16]  (WMMA f32 16x16x4, K=64)
// block = 128 threads = 4 waves; wave w -> row tile blockIdx.x*4 + w
// ---------------------------------------------------------------------------
__global__ __launch_bounds__(128) void gemm2_kernel(const float* __restrict__ act1,
                                                    const float* __restrict__ W2,
                                                    float* __restrict__ h2) {
    const int wave = threadIdx.x >> 5;
    const int lane = threadIdx.x & 31;
    const int half = lane >> 4;
    const int l    = lane & 15;
    const int tile = blockIdx.x * 4 + wave;
    const int m0   = tile * 16;
    if (m0 >= N_NODES) return;   // whole-wave early out; EXEC all-1s inside

    v8f c = {0.f, 0.f, 0.f, 0.f, 0.f, 0.f, 0.f, 0.f};
    const float* arow = act1 + (size_t)(m0 + l) * F1;

    for (int k0 = 0; k0 < F1; k0 += 4) {
        v2f a, b;
        const int ka = k0 + 2 * half;
        a.x = arow[ka];
        a.y = arow[ka + 1];
        b.x = W2[(size_t)ka * N_CLS + l];
        b.y = W2[(size_t)(ka + 1) * N_CLS + l];
        c = __builtin_amdgcn_wmma_f32_16x16x4_f32(false, a, false, b, (short)0, c,
                                                  false, false);
    }
#pragma unroll
    for (int i = 0; i < 8; ++i) {
        const int row = m0 + i + 8 * half;
        h2[(size_t)row * N_CLS + l] = c[i];
    }
}

__global__ void alphas2_kernel(const float* __restrict__ h2,
                               const float* __restrict__ a2s,
                               const float* __restrict__ a2d,
                               float* __restrict__ as, float* __restrict__ ad) {
    int n = blockIdx.x * blockDim.x + threadIdx.x;
    if (n >= N_NODES) return;
    const float* hr = h2 + (size_t)n * N_CLS;
    float ss = 0.f, sd = 0.f;
#pragma unroll
    for (int cidx = 0; cidx < N_CLS; ++cidx) {
        float v = hr[cidx];
        ss += v * a2s[cidx];
        sd += v * a2d[cidx];
    }
    as[n] = ss;
    ad[n] = sd;
}

// ---------------------------------------------------------------------------
// Edge passes, layer 2 (1 head, 16 dims)
// ---------------------------------------------------------------------------
__global__ void edge2_max(const int* __restrict__ src, const int* __restrict__ dst,
                          const float* __restrict__ as, const float* __restrict__ ad,
                          float* __restrict__ m2) {
    int stride = gridDim.x * blockDim.x;
    for (int e = blockIdx.x * blockDim.x + threadIdx.x; e < N_EDGES; e += stride) {
        int s = src[e], d = dst[e];
        atomicMaxF(&m2[d], leaky(as[s] + ad[d]));
    }
}

__global__ void edge2_den(const int* __restrict__ src, const int* __restrict__ dst,
                          const float* __restrict__ as, const float* __restrict__ ad,
                          const float* __restrict__ m2, float* __restrict__ den2) {
    int stride = gridDim.x * blockDim.x;
    for (int e = blockIdx.x * blockDim.x + threadIdx.x; e < N_EDGES; e += stride) {
        int s = src[e], d = dst[e];
        atomicAdd(&den2[d], expf(leaky(as[s] + ad[d]) - m2[d]));
    }
}

__global__ void edge2_agg(const int* __restrict__ src, const int* __restrict__ dst,
                          const float* __restrict__ as, const float* __restrict__ ad,
                          const float* __restrict__ m2, const float* __restrict__ den2,
                          const float* __restrict__ h2, float* __restrict__ out2) {
    int stride = gridDim.x * blockDim.x;
    for (int e = blockIdx.x * blockDim.x + threadIdx.x; e < N_EDGES; e += stride) {
        int s = src[e], d = dst[e];
        int en = e + stride;
        if (en < N_EDGES)
            __builtin_prefetch(&h2[(size_t)src[en] * N_CLS], 0, 0);
        float alpha = expf(leaky(as[s] + ad[d]) - m2[d]) / den2[d];
        const float4* hv = (const float4*)(h2 + (size_t)s * N_CLS);
        float* o = out2 + (size_t)d * N_CLS;
#pragma unroll
        for (int q = 0; q < 4; ++q) {
            float4 v = hv[q];
            atomicAdd(o + 4 * q + 0, alpha * v.x);
            atomicAdd(o + 4 * q + 1, alpha * v.y);
            atomicAdd(o + 4 * q + 2, alpha * v.z);
            atomicAdd(o + 4 * q + 3, alpha * v.w);
        }
    }
}

// out2 + b2, log_softmax over the 16 classes -> d_out
__global__ void final_kernel(const float* __restrict__ out2, const float* __restrict__ b2,
                             float* __restrict__ out) {
    int n = blockIdx.x * blockDim.x + threadIdx.x;
    if (n >= N_NODES) return;
    float v[N_CLS];
    float mx = -INFINITY;
#pragma unroll
    for (int cidx = 0; cidx < N_CLS; ++cidx) {
        v[cidx] = out2[(size_t)n * N_CLS + cidx] + b2[cidx];
        mx = fmaxf(mx, v[cidx]);
    }
    float ssum = 0.f;
#pragma unroll
    for (int cidx = 0; cidx < N_CLS; ++cidx) ssum += expf(v[cidx] - mx);
    float lg = logf(ssum) + mx;
#pragma unroll
    for (int cidx = 0; cidx < N_CLS; ++cidx)
        out[(size_t)n * N_CLS + cidx] = v[cidx] - lg;
}

// ---------------------------------------------------------------------------
extern "C" void kernel_launch(void* const* d_in, const int* in_sizes, int n_in,
                              void* d_out, int out_size, void* d_ws, size_t ws_size,
                              hipStream_t stream) {
    (void)in_sizes; (void)n_in; (void)out_size; (void)ws_size;

    const float* x    = (const float*)d_in[0];
    const int*   eidx = (const int*)d_in[1];
    const float* W1   = (const float*)d_in[2];
    const float* a1s  = (const float*)d_in[3];
    const float* a1d  = (const float*)d_in[4];
    const float* b1   = (const float*)d_in[5];
    const float* W2   = (const float*)d_in[6];
    const float* a2s  = (const float*)d_in[7];
    const float* a2d  = (const float*)d_in[8];
    const float* b2   = (const float*)d_in[9];
    float* out = (float*)d_out;

    const int* src = eidx;              // edge_index[0,:]
    const int* dst = eidx + N_EDGES;    // edge_index[1,:]

    // ---- carve workspace (all offsets 16B aligned: N_NODES*4 % 16 == 0) ----
    float* w = (float*)d_ws;
    float* h1   = w; w += (size_t)N_NODES * F1;      // 6.4M
    float* as1  = w; w += (size_t)N_NODES * HEADS;   // 0.8M
    float* ad1  = w; w += (size_t)N_NODES * HEADS;
    float* m1   = w; w += (size_t)N_NODES * HEADS;
    float* den1 = w; w += (size_t)N_NODES * HEADS;
    float* out1 = w; w += (size_t)N_NODES * F1;
    float* act1 = w; w += (size_t)N_NODES * F1;
    float* h2   = w; w += (size_t)N_NODES * N_CLS;
    float* as2  = w; w += (size_t)N_NODES;
    float* ad2  = w; w += (size_t)N_NODES;
    float* m2   = w; w += (size_t)N_NODES;
    float* den2 = w; w += (size_t)N_NODES;
    float* out2 = w; w += (size_t)N_NODES * N_CLS;

    const int EB = 256;                       // edge-pass block
    const int EG = 4096;                      // edge-pass grid (grid-stride)
    const int NB = (N_NODES + 255) / 256;     // per-node grids
    const int T1 = N_NODES / 16;              // 6250 row tiles (exact)

    init_kernel<<<1024, 256, 0, stream>>>(m1, den1, out1, m2, den2, out2);

    // Layer 1
    gemm1_kernel<<<T1, 128, 0, stream>>>(x, W1, h1);
    alphas1_kernel<<<NB, 256, 0, stream>>>(h1, a1s, a1d, as1, ad1);
    edge1_max<<<EG, EB, 0, stream>>>(src, dst, as1, ad1, m1);
    edge1_den<<<EG, EB, 0, stream>>>(src, dst, as1, ad1, m1, den1);
    edge1_agg<<<EG, EB, 0, stream>>>(src, dst, as1, ad1, m1, den1, h1, out1);
    elu_kernel<<<2048, 256, 0, stream>>>(out1, b1, act1);

    // Layer 2
    gemm2_kernel<<<(T1 + 3) / 4, 128, 0, stream>>>(act1, W2, h2);
    alphas2_kernel<<<NB, 256, 0, stream>>>(h2, a2s, a2d, as2, ad2);
    edge2_max<<<EG, EB, 0, stream>>>(src, dst, as2, ad2, m2);
    edge2_den<<<EG, EB, 0, stream>>>(src, dst, as2, ad2, m2, den2);
    edge2_agg<<<EG, EB, 0, stream>>>(src, dst, as2, ad2, m2, den2, h2, out2);
    final_kernel<<<NB, 256, 0, stream>>>(out2, b2, out);
}